// PaiConvDG_91285234909707
// MI455X (gfx1250) — compile-verified
//
#include <hip/hip_runtime.h>
#include <math.h>

typedef __attribute__((ext_vector_type(2))) float v2f;
typedef __attribute__((ext_vector_type(8))) float v8f;

#define BQ    16
#define CQ    64
#define NQ    2048
#define NPTS  (BQ * NQ)     // 32768 points
#define KN    20            // neighbors
#define KSQ   20            // permatrix cols
#define OCQ   64            // out channels
#define TWO_C 128
#define WSTRIDE 72          // LDS row stride (words) for W stack  -> disjoint bank halves
#define GSTRIDE 72          // LDS row stride (words) for G' slab
#define BN_EPS 1e-5f

// D = A(16x4 f32) * B(4x16 f32) + C(16x16 f32), wave32 WMMA
__device__ __forceinline__ v8f wmma4(v2f a, v2f b, v8f c) {
    return __builtin_amdgcn_wmma_f32_16x16x4_f32(
        /*neg_a=*/false, a, /*neg_b=*/false, b,
        /*c_mod=*/(short)0, c, /*reuse_a=*/false, /*reuse_b=*/false);
}

// -------- Kernel 0: feature [B,C,N] -> flatF [B*N, C]  (+ one zero pad row) --------
__global__ void prep_transpose(const float* __restrict__ feat,
                               float* __restrict__ flatF) {
    int gid = blockIdx.x * 256 + threadIdx.x;
    if (gid >= NPTS * CQ + CQ) return;
    if (gid >= NPTS * CQ) {          // zero row: source for padded A-fragment lanes
        flatF[gid] = 0.f;
        return;
    }
    int r = gid >> 6;                // point index b*N+n
    int c = gid & 63;
    int b = r >> 11;
    int n = r & (NQ - 1);
    flatF[gid] = feat[((b * CQ + c) << 11) + n];   // coalesced writes, L2-resident reads
}

// ---------------- Kernel 1: fused gather + GEMM1 + GEMM2 + max ----------------
__global__ __launch_bounds__(128) void paiconv_main(
    const float* __restrict__ flatF,    // [NPTS+1, CQ] (last row zeros)
    const int*   __restrict__ nidx,     // [NPTS, KN]
    const float* __restrict__ perm,     // [NPTS, KN, KSQ]
    const float* __restrict__ conv_w,   // [OCQ, 2C]
    const float* __restrict__ conv_b,   // [OCQ]
    float* __restrict__ out_pre,        // [NPTS, OCQ]
    float* __restrict__ gsum,           // [OCQ]
    float* __restrict__ gsumsq)         // [OCQ]
{
    __shared__ float sW[TWO_C * WSTRIDE];        // stacked [W2^T ; (W1-W2)^T] as [c][o]
    __shared__ float sG[4][KN * GSTRIDE];        // per-wave G' [k][o]
    __shared__ float sSum[OCQ], sSumSq[OCQ];

    const int tid  = threadIdx.x;
    const int wave = tid >> 5;
    const int lane = tid & 31;
    const int half = lane >> 4;      // lane-half selects K offset
    const int lm   = lane & 15;

    // Stage W stack: sW[c][o]; c<64 -> W2^T = conv_w[o][64+c]; c>=64 -> (W1-W2)^T
    for (int i = tid; i < TWO_C * OCQ; i += 128) {
        int c = i >> 6, o = i & 63;
        float w2 = conv_w[o * TWO_C + 64 + (c & 63)];
        float v  = (c < 64) ? w2 : (conv_w[o * TWO_C + (c - 64)] - w2);
        sW[c * WSTRIDE + o] = v;
    }
    if (tid < OCQ) { sSum[tid] = 0.f; sSumSq[tid] = 0.f; }
    __syncthreads();

    const int p = blockIdx.x * 4 + wave;
    float* myG = sG[wave];

    // Per-lane neighbor row indices for A fragments (A row m = neighbor index).
    // Padded lanes (m >= KN) use row index NPTS = zero row: no masking in hot loop,
    // and all addresses stay single global-addrspace expressions off flatF
    // (integer select, NOT pointer select -> global_load_b64, not flat_load).
    const int idx0 = nidx[p * KN];
    long offA[2], offU[2];
    for (int mt = 0; mt < 2; ++mt) {
        int m = mt * 16 + lm;
        int ia = (m < KN) ? nidx[p * KN + m] : NPTS;   // gather half   (c < 64)
        int iu = (m < KN) ? idx0             : NPTS;   // f0 broadcast  (c >= 64)
        offA[mt] = (long)ia * CQ;
        offU[mt] = (long)iu * CQ;
    }
    const float* Pp = perm + (long)p * (KN * KSQ);
    __builtin_prefetch(Pp, 0, 0);                    // global_prefetch_b8: warm permatrix

    // ---- GEMM1: G'^T(20x64,pad32) = [F^T | f0^T](20x128) x [W2^T;Wd^T](128x64) ----
    v8f acc[2][4] = {};
    for (int kc = 0; kc < TWO_C; kc += 4) {
        const int cA = kc + 2 * half;                // this lane's K pair {cA, cA+1}
        v2f a[2];
        if (kc < 64) {
            a[0] = *(const v2f*)(flatF + offA[0] + cA);
            a[1] = *(const v2f*)(flatF + offA[1] + cA);
        } else {
            a[0] = *(const v2f*)(flatF + offU[0] + (cA - 64));
            a[1] = *(const v2f*)(flatF + offU[1] + (cA - 64));
        }
        for (int nt = 0; nt < 4; ++nt) {
            int o = nt * 16 + lm;
            v2f b;
            b.x = sW[(cA    ) * WSTRIDE + o];
            b.y = sW[(cA + 1) * WSTRIDE + o];
            acc[0][nt] = wmma4(a[0], b, acc[0][nt]);
            acc[1][nt] = wmma4(a[1], b, acc[1][nt]);
        }
    }

    // Spill G' (only rows k<20 are ever re-read) to this wave's LDS slab
    for (int mt = 0; mt < 2; ++mt)
        for (int nt = 0; nt < 4; ++nt)
            for (int v = 0; v < 8; ++v) {
                int k = mt * 16 + v + 8 * half;      // C-layout row
                if (k < KN) myG[k * GSTRIDE + nt * 16 + lm] = acc[mt][nt][v];
            }

    // ---- GEMM2: M^T(20x64) = P^T(20x20) x G'(20x64) ----
    v8f acc2[2][4] = {};
    for (int kc = 0; kc < KN; kc += 4) {             // kc = 0,4,8,12,16 ; rows <= 19
        const int kk = kc + 2 * half;
        v2f pa[2];
        for (int mt = 0; mt < 2; ++mt) {
            int s = mt * 16 + lm;
            float x0 = 0.f, x1 = 0.f;
            if (s < KSQ) {                           // coalesced: consecutive lanes -> consecutive s
                x0 = Pp[kk * KSQ + s];
                x1 = Pp[(kk + 1) * KSQ + s];
            }
            pa[mt].x = x0; pa[mt].y = x1;
        }
        for (int nt = 0; nt < 4; ++nt) {
            v2f b;
            b.x = myG[(kk    ) * GSTRIDE + nt * 16 + lm];
            b.y = myG[(kk + 1) * GSTRIDE + nt * 16 + lm];
            acc2[0][nt] = wmma4(pa[0], b, acc2[0][nt]);
            acc2[1][nt] = wmma4(pa[1], b, acc2[1][nt]);
        }
    }

    // ---- Epilogue: max over s (20), + bias, stats, store pre-BN ----
    for (int nt = 0; nt < 4; ++nt) {
        int o = nt * 16 + lm;
        float mx = -3.402823466e+38f;
        for (int v = 0; v < 8; ++v) {
            mx = fmaxf(mx, acc2[0][nt][v]);          // s = v+8*half in [0,16): always valid
            int s1 = 16 + v + 8 * half;
            float c1 = acc2[1][nt][v];
            mx = (s1 < KSQ) ? fmaxf(mx, c1) : mx;
        }
        mx = fmaxf(mx, __shfl_xor(mx, 16, 32));      // merge lane halves (same o)
        float val = mx + conv_b[o];
        if (half == 0) {
            out_pre[(long)p * OCQ + o] = val;        // 64B coalesced per n-tile
            atomicAdd(&sSum[o], val);                // ds_add_f32
            atomicAdd(&sSumSq[o], val * val);
        }
    }
    __syncthreads();
    if (tid < OCQ)        atomicAdd(&gsum[tid], sSum[tid]);
    else if (tid < 2*OCQ) atomicAdd(&gsumsq[tid - OCQ], sSumSq[tid - OCQ]);
}

// ---------------- Kernel 2: BatchNorm finalize, write [B, OC, N] ----------------
__global__ void bn_finalize(const float* __restrict__ out_pre,
                            const float* __restrict__ gsum,
                            const float* __restrict__ gsumsq,
                            const float* __restrict__ gamma,
                            const float* __restrict__ beta,
                            float* __restrict__ out) {
    int gid = blockIdx.x * 256 + threadIdx.x;
    if (gid >= NPTS * OCQ) return;
    int n = gid & (NQ - 1);
    int o = (gid >> 11) & 63;
    int b = gid >> 17;
    float x = out_pre[(long)(b * NQ + n) * OCQ + o];
    const float invN = 1.0f / (float)NPTS;
    float mean = gsum[o] * invN;
    float var  = gsumsq[o] * invN - mean * mean;     // biased, matches jnp.var
    float y = (x - mean) * rsqrtf(var + BN_EPS);
    out[gid] = y * gamma[o] + beta[o];
}

extern "C" void kernel_launch(void* const* d_in, const int* in_sizes, int n_in,
                              void* d_out, int out_size, void* d_ws, size_t ws_size,
                              hipStream_t stream) {
    const float* feature = (const float*)d_in[0];
    const int*   nidx    = (const int*)  d_in[1];
    const float* perm    = (const float*)d_in[2];
    const float* conv_w  = (const float*)d_in[3];
    const float* conv_b  = (const float*)d_in[4];
    const float* gamma   = (const float*)d_in[5];
    const float* beta    = (const float*)d_in[6];
    float* out = (float*)d_out;

    float* flatF   = (float*)d_ws;                           // 8 MB + 256 B zero row
    float* out_pre = flatF + (size_t)(NPTS + 1) * CQ;        // 8 MB
    float* gsum    = out_pre + (size_t)NPTS * OCQ;           // 256 B
    float* gsumsq  = gsum + OCQ;                             // 256 B

    hipMemsetAsync(gsum, 0, 2 * OCQ * sizeof(float), stream);

    prep_transpose<<<(NPTS * CQ + CQ + 255) / 256, 256, 0, stream>>>(feature, flatF);
    paiconv_main<<<NPTS / 4, 128, 0, stream>>>(flatF, nidx, perm, conv_w, conv_b,
                                               out_pre, gsum, gsumsq);
    bn_finalize<<<(NPTS * OCQ + 255) / 256, 256, 0, stream>>>(out_pre, gsum, gsumsq,
                                                              gamma, beta, out);
}